// OpSolve_71004399338137
// MI455X (gfx1250) — compile-verified
//
#include <hip/hip_runtime.h>
#include <hip/hip_bf16.h>

#define VDD_F   1.0f
#define TOL_F   1e-3f
#define MAXIT   50

// ---------------------------------------------------------------------------
// Setup: j = -current ; zero X, Y, invd (deg accumulator), dist slots, stop.
// ---------------------------------------------------------------------------
__global__ void k_setup(const float* __restrict__ current,
                        float* __restrict__ X, float* __restrict__ Y,
                        float* __restrict__ invd, float* __restrict__ jv,
                        float* __restrict__ dist, int* __restrict__ stop, int N) {
    int i = blockIdx.x * blockDim.x + threadIdx.x;
    if (i < N) {
        X[i] = 0.0f;
        Y[i] = 0.0f;
        invd[i] = 0.0f;          // will accumulate deg
        jv[i] = -current[i];
    }
    if (i < 64) dist[i] = 0.0f;
    if (i == 0) *stop = 0;
}

// ---------------------------------------------------------------------------
// deg[row] += w   (segment_sum of edge weights by row)
// ---------------------------------------------------------------------------
__global__ void k_deg(const int* __restrict__ row, const float* __restrict__ w,
                      float* __restrict__ deg, int E) {
    int t = blockIdx.x * blockDim.x + threadIdx.x;
    int base = t * 4;
    if (base + 3 < E) {
        int4   r  = *(const int4*)(row + base);
        float4 ww = *(const float4*)(w + base);
        atomicAdd(&deg[r.x], ww.x);
        atomicAdd(&deg[r.y], ww.y);
        atomicAdd(&deg[r.z], ww.z);
        atomicAdd(&deg[r.w], ww.w);
    } else {
        for (int e = base; e < E; ++e) atomicAdd(&deg[row[e]], w[e]);
    }
}

// ---------------------------------------------------------------------------
// Candidate nodes: d += x ; j += VDD*x   (can_index is unique -> plain RMW ok)
// ---------------------------------------------------------------------------
__global__ void k_can(const float* __restrict__ x, const int* __restrict__ can,
                      float* __restrict__ dvec, float* __restrict__ jv, int NC) {
    int t = blockIdx.x * blockDim.x + threadIdx.x;
    if (t < NC) {
        int idx = can[t];
        dvec[idx] = dvec[idx] + x[t];
        jv[idx]   = jv[idx]   + VDD_F * x[t];
    }
}

__global__ void k_recip(float* __restrict__ dvec, int N) {
    int i = blockIdx.x * blockDim.x + threadIdx.x;
    if (i < N) dvec[i] = 1.0f / dvec[i];
}

// ---------------------------------------------------------------------------
// SpMV scatter: acc[row] += w * V[col]   (COO, atomics; acc pre-zeroed).
// Edge streams are L2-resident on MI455X (144MB < 192MB L2); prefetch ahead
// via the gfx1250 global_prefetch_b8 path.
// ---------------------------------------------------------------------------
__global__ void k_spmv(const int* __restrict__ row, const int* __restrict__ col,
                       const float* __restrict__ w, const float* __restrict__ V,
                       float* __restrict__ acc, const int* __restrict__ stop, int E) {
    if (*stop) return;   // converged: loop body suppressed (uniform branch)
    int t = blockIdx.x * blockDim.x + threadIdx.x;
    int base = t * 4;
    if (base + 3 < E) {
        __builtin_prefetch(row + base + 4096, 0, 0);
        __builtin_prefetch(col + base + 4096, 0, 0);
        __builtin_prefetch(w   + base + 4096, 0, 0);
        int4   r  = *(const int4*)(row + base);
        int4   c  = *(const int4*)(col + base);
        float4 ww = *(const float4*)(w + base);
        atomicAdd(&acc[r.x], ww.x * V[c.x]);
        atomicAdd(&acc[r.y], ww.y * V[c.y]);
        atomicAdd(&acc[r.z], ww.z * V[c.z]);
        atomicAdd(&acc[r.w], ww.w * V[c.w]);
    } else {
        for (int e = base; e < E; ++e) atomicAdd(&acc[row[e]], w[e] * V[col[e]]);
    }
}

// ---------------------------------------------------------------------------
// Fused Jacobi step: dst holds acc = A*Vcur.  dst = (acc + j) * invd,
// dist_slot += sum |dst - src|, then src is zeroed (accumulator prep for the
// next iteration).  If converged: copy-through dst=src, zero src (keeps the
// final buffer holding the last iterate, as while_loop exit semantics).
// wave32 shuffle + LDS block reduction.
// ---------------------------------------------------------------------------
__global__ void k_jacobi(float* __restrict__ dst, float* __restrict__ src,
                         const float* __restrict__ jv, const float* __restrict__ invd,
                         float* __restrict__ dist_slot, const int* __restrict__ stop,
                         int N) {
    __shared__ float sred[8];
    const int tid = blockIdx.x * blockDim.x + threadIdx.x;
    const int i4  = tid * 4;
    const bool stopped = (*stop != 0);

    if (stopped) {
        if (i4 + 3 < N) {
            float4 v = *(const float4*)(src + i4);
            *(float4*)(dst + i4) = v;
            float4 z; z.x = 0.f; z.y = 0.f; z.z = 0.f; z.w = 0.f;
            *(float4*)(src + i4) = z;
        } else {
            for (int k = i4; k < N; ++k) { dst[k] = src[k]; src[k] = 0.0f; }
        }
        return;  // uniform across block: no barrier divergence
    }

    float local = 0.0f;
    if (i4 + 3 < N) {
        float4 a  = *(const float4*)(dst + i4);
        float4 jj = *(const float4*)(jv + i4);
        float4 dd = *(const float4*)(invd + i4);
        float4 vc = *(const float4*)(src + i4);
        float4 vn;
        vn.x = (a.x + jj.x) * dd.x;
        vn.y = (a.y + jj.y) * dd.y;
        vn.z = (a.z + jj.z) * dd.z;
        vn.w = (a.w + jj.w) * dd.w;
        local = fabsf(vn.x - vc.x) + fabsf(vn.y - vc.y) +
                fabsf(vn.z - vc.z) + fabsf(vn.w - vc.w);
        *(float4*)(dst + i4) = vn;
        float4 z; z.x = 0.f; z.y = 0.f; z.z = 0.f; z.w = 0.f;
        *(float4*)(src + i4) = z;
    } else {
        for (int k = i4; k < N; ++k) {
            float vn = (dst[k] + jv[k]) * invd[k];
            local += fabsf(vn - src[k]);
            dst[k] = vn;
            src[k] = 0.0f;
        }
    }

    // wave32 reduction
    for (int off = 16; off > 0; off >>= 1)
        local += __shfl_down(local, off, 32);
    const int lane = threadIdx.x & 31;
    const int wid  = threadIdx.x >> 5;
    if (lane == 0) sred[wid] = local;
    __syncthreads();
    if (wid == 0) {
        float s = (lane < (int)(blockDim.x >> 5)) ? sred[lane] : 0.0f;
        for (int off = 4; off > 0; off >>= 1)
            s += __shfl_down(s, off, 32);
        if (lane == 0) atomicAdd(dist_slot, s);
    }
}

// while_loop gate: once dist <= TOL, stop sticks (later slots are 0 <= TOL).
__global__ void k_flag(const float* __restrict__ dist_slot, int* __restrict__ stop) {
    if (*dist_slot <= TOL_F) *stop = 1;
}

__global__ void k_gather(const float* __restrict__ V, const int* __restrict__ obs,
                         float* __restrict__ out, int M) {
    int t = blockIdx.x * blockDim.x + threadIdx.x;
    if (t < M) out[t] = V[obs[t]];
}

// ---------------------------------------------------------------------------
extern "C" void kernel_launch(void* const* d_in, const int* in_sizes, int n_in,
                              void* d_out, int out_size, void* d_ws, size_t ws_size,
                              hipStream_t stream) {
    const float* x       = (const float*)d_in[0];
    const int*   erow    = (const int*)  d_in[1];
    const int*   ecol    = (const int*)  d_in[2];
    const float* ew      = (const float*)d_in[3];
    const float* current = (const float*)d_in[4];
    const int*   can     = (const int*)  d_in[5];
    const int*   obs     = (const int*)  d_in[6];
    float*       out     = (float*)d_out;

    const int NC   = in_sizes[0];
    const int E    = in_sizes[1];
    const int N    = in_sizes[4];
    const int NOBS = in_sizes[6];

    float* ws   = (float*)d_ws;
    float* X    = ws;                 // ping buffer (starts as V0 = 0)
    float* Y    = ws + (size_t)N;     // pong buffer
    float* invd = ws + 2 * (size_t)N;
    float* jv   = ws + 3 * (size_t)N;
    float* dist = ws + 4 * (size_t)N; // 64 per-iteration L1-distance slots
    int*   stop = (int*)(dist + 64);

    const int B = 256;
    const int gN  = (N + B - 1) / B;            // 1 elem / thread
    const int gN4 = (N + 4 * B - 1) / (4 * B);  // 4 elems / thread
    const int gE4 = (E + 4 * B - 1) / (4 * B);  // 4 edges / thread
    const int gC  = (NC + B - 1) / B;
    const int gO  = (NOBS + B - 1) / B;

    // --- setup: j, d, invd ---
    k_setup<<<gN, B, 0, stream>>>(current, X, Y, invd, jv, dist, stop, N);
    k_deg  <<<gE4, B, 0, stream>>>(erow, ew, invd, E);
    k_can  <<<gC, B, 0, stream>>>(x, can, invd, jv, NC);
    k_recip<<<gN, B, 0, stream>>>(invd, N);

    // --- iteration 0: V1 = (0 + j)/d  (Y was zeroed: acc == 0) ---
    k_jacobi<<<gN4, B, 0, stream>>>(Y, X, jv, invd, dist + 0, stop, N);
    k_flag  <<<1, 1, 0, stream>>>(dist + 0, stop);

    float* cur = Y;
    float* nxt = X;   // invariant: nxt is zeroed before each spmv

    for (int k = 1; k <= MAXIT; ++k) {
        k_spmv  <<<gE4, B, 0, stream>>>(erow, ecol, ew, cur, nxt, stop, E);
        k_jacobi<<<gN4, B, 0, stream>>>(nxt, cur, jv, invd, dist + k, stop, N);
        if (k < MAXIT) k_flag<<<1, 1, 0, stream>>>(dist + k, stop);
        float* t = cur; cur = nxt; nxt = t;
    }

    k_gather<<<gO, B, 0, stream>>>(cur, obs, out, NOBS);
}